// LSDGINNet_22574348108064
// MI455X (gfx1250) — compile-verified
//
#include <hip/hip_runtime.h>
#include <stdint.h>

// ---------------------------------------------------------------------------
// GIN-style GNN forward for MI455X (gfx1250, wave32).
//   pre1 = [agg(x,ei) + x , agg(x,eo) + x]            [N,256]
//   h    = relu(pre1@W1+b1) @ W2 + b2                 [N,256]
//   pre2 = [agg(h,ei) + h , agg(h,eo) + h]            [N,512]
//   out  = pre2 @ W3 + b3                             [N,128]
// GEMMs: V_WMMA_F32_16X16X4_F32, LDS-tiled with async global->LDS copies
// (ASYNCcnt) and double buffering.
// ---------------------------------------------------------------------------

#define N_NODES 50000
#define N_EDGES 800000
#define IN_DIM  128
#define HID_DIM 256
#define OUT_DIM 128

typedef __attribute__((ext_vector_type(2))) float v2f;
typedef __attribute__((ext_vector_type(8))) float v8f;
typedef __attribute__((address_space(3))) float lds_f;

#define BM 64
#define BN 64
#define BK 32
#define APAD 4
#define ASTRIDE (BK + APAD)     /* 36 floats: conflict-free A-frag ds_load_b64 */

// ---------------------------------------------------------------------------
// CDNA5 async copy: 16B global -> LDS, tracked by ASYNCcnt (no VGPR round trip)
// ---------------------------------------------------------------------------
__device__ __forceinline__ void async_b128(float* lds_ptr, const float* gptr) {
    lds_f* l = (lds_f*)lds_ptr;                       // flat -> LDS addrspace
    asm volatile("global_load_async_to_lds_b128 %0, %1, off"
                 :: "v"(l), "v"(gptr) : "memory");
}
__device__ __forceinline__ void wait_async0() {
    asm volatile("s_wait_asynccnt 0x0" ::: "memory");
}

// ---------------------------------------------------------------------------
// pre[i, 0:D] = h[i];  pre[i, D:2D] = h[i]   ((1+eps)*cat([h,h]), eps=0)
// ---------------------------------------------------------------------------
__global__ void fanout_kernel(const float4* __restrict__ src,
                              float4* __restrict__ dst,
                              int n, int d4) {
    long long i = (long long)blockIdx.x * blockDim.x + threadIdx.x;
    long long total = (long long)n * d4;
    if (i >= total) return;
    long long node = i / d4;
    int f = (int)(i - node * d4);
    float4 v = src[i];
    float4* row = dst + node * (2LL * d4);
    row[f]      = v;
    row[d4 + f] = v;
}

// ---------------------------------------------------------------------------
// Scatter-add: pre[dst, colOff + 0:D] += h[src, 0:D].
// Wave-uniform edge index, coalesced float4 gathers, fp32 atomics in L2.
// ---------------------------------------------------------------------------
__global__ void scatter_add_kernel(const float* __restrict__ h,
                                   const long long* __restrict__ esrc,
                                   const long long* __restrict__ edst,
                                   float* __restrict__ pre,
                                   long long n_edges, int D, int ld, int colOff) {
    long long t = (long long)blockIdx.x * blockDim.x + threadIdx.x;
    int d4 = D >> 2;
    long long total = n_edges * d4;
    if (t >= total) return;
    long long e = t / d4;
    int f = (int)(t - e * d4) << 2;

    long long s = esrc[e];
    long long d = edst[e];

    const float4 v = *(const float4*)(h + s * (long long)D + f);
    float* p = pre + d * (long long)ld + colOff + f;
    atomicAdd(p + 0, v.x);
    atomicAdd(p + 1, v.y);
    atomicAdd(p + 2, v.z);
    atomicAdd(p + 3, v.w);
}

// ---------------------------------------------------------------------------
// Stage one BMxBK A-tile and BKxBN B-tile into LDS via async b128 copies.
// 256 threads: A = 64 rows x 8 float4 (4 thr/row, 2 chunks each);
//              B = 32 rows x 16 float4 (8 thr/row, 2 chunks each).
// ---------------------------------------------------------------------------
__device__ __forceinline__ void stage_tiles(const float* __restrict__ A,
                                            const float* __restrict__ B,
                                            float* Asm, float* Bsm,
                                            int row0, int colB0, int kc,
                                            int M, int N, int K,
                                            int ar, int aseg, int br, int bseg) {
    int arow = row0 + ar;
    if (arow >= M) arow = M - 1;                 // clamp OOB gather rows
    const float* ga = A + (long long)arow * K + kc;
    async_b128(Asm + ar * ASTRIDE + aseg,      ga + aseg);
    async_b128(Asm + ar * ASTRIDE + aseg + 16, ga + aseg + 16);

    const float* gb = B + (long long)(kc + br) * N + colB0;
    async_b128(Bsm + br * BN + bseg,      gb + bseg);
    async_b128(Bsm + br * BN + bseg + 32, gb + bseg + 32);
}

// ---------------------------------------------------------------------------
// C[M,N] = act(A[M,K] @ B[K,N] + bias).  Block 256 thr = 8 waves (4x2).
// Each wave: 16x32 output (two 16x16 f32 accumulators), K stepped by 4 with
// V_WMMA_F32_16X16X4_F32. Double-buffered LDS tiles, async prefetch.
//
// Fragment layouts (ISA 7.12.2, wave32):
//   A 16x4 : m = lane%16 ; v0=A[m][kk], v1=A[m][kk+1], kk = k + 2*(lane>=16)
//   B 4x16 : n = lane%16 ; v0=B[kk][n], v1=B[kk+1][n]
//   C 16x16: reg r -> row r + 8*(lane>=16), col lane%16
// ---------------------------------------------------------------------------
__global__ void __launch_bounds__(256)
wmma_gemm_kernel(const float* __restrict__ A,
                 const float* __restrict__ B,
                 const float* __restrict__ bias,
                 float* __restrict__ C,
                 int M, int N, int K, int relu) {
    __shared__ float Asm[2][BM * ASTRIDE];
    __shared__ float Bsm[2][BK * BN];

    const int t    = threadIdx.x;
    const int wave = t >> 5;
    const int lane = t & 31;
    const int half = lane >> 4;          // K-pair select
    const int l    = lane & 15;          // A row / B col within tile
    const int wrow = wave >> 1;          // 0..3
    const int wcol = wave & 1;           // 0..1

    const int row0  = blockIdx.y * BM;
    const int colB0 = blockIdx.x * BN;

    const int ar = t >> 2, aseg = (t & 3) << 2;   // A staging role
    const int br = t >> 3, bseg = (t & 7) << 2;   // B staging role

    v8f acc0 = {};
    v8f acc1 = {};

    stage_tiles(A, B, Asm[0], Bsm[0], row0, colB0, 0, M, N, K, ar, aseg, br, bseg);
    wait_async0();
    __syncthreads();

    int buf = 0;
    for (int kc = 0; kc < K; kc += BK) {
        if (kc + BK < K)                 // prefetch next chunk into other buffer
            stage_tiles(A, B, Asm[buf ^ 1], Bsm[buf ^ 1], row0, colB0, kc + BK,
                        M, N, K, ar, aseg, br, bseg);

        const float* As = Asm[buf] + (wrow * 16 + l) * ASTRIDE;
        const float* Bs = Bsm[buf] + wcol * 32 + l;

#pragma unroll
        for (int k4 = 0; k4 < BK; k4 += 4) {
            const int kk = k4 + 2 * half;
            v2f a = *(const v2f*)(As + kk);                    // ds_load_b64
            v2f b0; b0.x = Bs[kk * BN];      b0.y = Bs[(kk + 1) * BN];
            acc0 = __builtin_amdgcn_wmma_f32_16x16x4_f32(
                false, a, false, b0, (short)0, acc0, false, false);
            v2f b1; b1.x = Bs[kk * BN + 16]; b1.y = Bs[(kk + 1) * BN + 16];
            acc1 = __builtin_amdgcn_wmma_f32_16x16x4_f32(
                false, a, false, b1, (short)0, acc1, false, false);
        }

        wait_async0();                   // own prefetch done
        __syncthreads();                 // everyone done reading buf / writing buf^1
        buf ^= 1;
    }

    // ---- epilogue: bias (+ReLU), predicated on row < M ----
    const int c0  = colB0 + wcol * 32 + l;
    const float bv0 = bias[c0];
    const float bv1 = bias[c0 + 16];
#pragma unroll
    for (int r = 0; r < 8; ++r) {
        const int row = row0 + wrow * 16 + r + 8 * half;
        if (row < M) {
            float v0 = acc0[r] + bv0;
            float v1 = acc1[r] + bv1;
            if (relu) { v0 = v0 > 0.0f ? v0 : 0.0f; v1 = v1 > 0.0f ? v1 : 0.0f; }
            C[(long long)row * N + c0]      = v0;
            C[(long long)row * N + c0 + 16] = v1;
        }
    }
}

// ---------------------------------------------------------------------------
// Host-side orchestration (graph-capture safe: launches only).
// ---------------------------------------------------------------------------
extern "C" void kernel_launch(void* const* d_in, const int* in_sizes, int n_in,
                              void* d_out, int out_size, void* d_ws, size_t ws_size,
                              hipStream_t stream) {
    const float*     x   = (const float*)d_in[0];   // [N, 128]
    const float*     W1  = (const float*)d_in[1];   // [256, 256]
    const float*     b1  = (const float*)d_in[2];   // [256]
    const float*     W2  = (const float*)d_in[3];   // [256, 256]
    const float*     b2  = (const float*)d_in[4];   // [256]
    const float*     W3  = (const float*)d_in[5];   // [512, 128]
    const float*     b3  = (const float*)d_in[6];   // [128]
    const long long* ei  = (const long long*)d_in[7];  // [2, E] int64
    const long long* eo  = (const long long*)d_in[8];  // [2, E] int64

    float* out = (float*)d_out;                     // [N, 128]

    // Workspace (aliased; peak = N*768 floats ~= 154 MB):
    //   pre1 : ws[0     , N*256)   dead after gemm1
    //   h1   : ws[N*256 , N*512)   dead after gemm2
    //   h2   : ws[0     , N*256)   reuses pre1 slot
    //   pre2 : ws[N*256 , N*768)   reuses h1 slot (+ extension)
    float* ws   = (float*)d_ws;
    float* pre1 = ws;
    float* h1   = ws + (long long)N_NODES * 256;
    float* h2   = ws;
    float* pre2 = ws + (long long)N_NODES * 256;

    const long long E = N_EDGES;
    const int TB = 256;
    const unsigned gy = (N_NODES + BM - 1) / BM;    // 782

    // ---------------- Layer 1: aggregate x (D=128) ----------------
    {
        long long total = (long long)N_NODES * (IN_DIM / 4);
        fanout_kernel<<<(unsigned)((total + TB - 1) / TB), TB, 0, stream>>>(
            (const float4*)x, (float4*)pre1, N_NODES, IN_DIM / 4);

        long long st = E * (IN_DIM / 4);
        unsigned sg = (unsigned)((st + TB - 1) / TB);
        scatter_add_kernel<<<sg, TB, 0, stream>>>(x, ei, ei + E, pre1, E, IN_DIM, 2 * IN_DIM, 0);
        scatter_add_kernel<<<sg, TB, 0, stream>>>(x, eo, eo + E, pre1, E, IN_DIM, 2 * IN_DIM, IN_DIM);
    }

    // ---------------- MLP: h = relu(pre1@W1+b1)@W2+b2 ----------------
    {
        dim3 grid(HID_DIM / BN, gy);                 // (4, 782)
        wmma_gemm_kernel<<<grid, 256, 0, stream>>>(pre1, W1, b1, h1,
                                                   N_NODES, HID_DIM, 2 * IN_DIM, 1);
        wmma_gemm_kernel<<<grid, 256, 0, stream>>>(h1, W2, b2, h2,
                                                   N_NODES, HID_DIM, HID_DIM, 0);
    }

    // ---------------- Layer 2: aggregate h (D=256) ----------------
    {
        long long total = (long long)N_NODES * (HID_DIM / 4);
        fanout_kernel<<<(unsigned)((total + TB - 1) / TB), TB, 0, stream>>>(
            (const float4*)h2, (float4*)pre2, N_NODES, HID_DIM / 4);

        long long st = E * (HID_DIM / 4);
        unsigned sg = (unsigned)((st + TB - 1) / TB);
        scatter_add_kernel<<<sg, TB, 0, stream>>>(h2, ei, ei + E, pre2, E, HID_DIM, 2 * HID_DIM, 0);
        scatter_add_kernel<<<sg, TB, 0, stream>>>(h2, eo, eo + E, pre2, E, HID_DIM, 2 * HID_DIM, HID_DIM);
    }

    // ---------------- Output: out = pre2@W3 + b3 ----------------
    {
        dim3 grid(OUT_DIM / BN, gy);                 // (2, 782)
        wmma_gemm_kernel<<<grid, 256, 0, stream>>>(pre2, W3, b3, out,
                                                   N_NODES, OUT_DIM, 2 * HID_DIM, 0);
    }
}